// MultiPartLoss_27805618274976
// MI455X (gfx1250) — compile-verified
//
#include <hip/hip_runtime.h>
#include <stdint.h>
#include <math.h>

// Problem constants (from reference): S=7, B=2, C=20, N_BATCH=32768
#define FEAT        30                    // B*5 + C floats per cell
#define NBATCH      32768
#define GCELLS      49                    // S*S
#define MCELLS      (NBATCH * GCELLS)     // 1,605,632 cells
#define TILE_CELLS  256
#define NTILES      (MCELLS / TILE_CELLS) // 6272 (exact)
#define TILE_FLOATS (TILE_CELLS * FEAT)   // 7680
#define TILE_BYTES  (TILE_FLOATS * 4)     // 30720
#define CHUNKS      (TILE_BYTES / 16)     // 1920 b128 chunks per array
#define NBLOCKS     1024
#define NTHREADS    256
#define NOOBJ_W     0.5f

__device__ __forceinline__ float iou_f(float px, float py, float pw, float ph,
                                       float tx, float ty, float tw, float th) {
    float xA = fmaxf(px - pw * 0.5f, tx - tw * 0.5f);
    float yA = fmaxf(py - ph * 0.5f, ty - th * 0.5f);
    float xB = fminf(px + pw * 0.5f, tx + tw * 0.5f);
    float yB = fminf(py + ph * 0.5f, ty + th * 0.5f);
    float inter = fmaxf(0.0f, xB - xA + 1.0f) * fmaxf(0.0f, yB - yA + 1.0f);
    float uni = pw * ph + tw * th - inter;
    return inter / uni;
}

// Issue 15 coalesced 16B async DMA chunks for one tile (preds half + targets
// half; the split at chunk 1920 falls on a wave boundary -> wave-uniform).
// Non-temporal: 385 MB streamed once > 192 MB L2, zero reuse per chunk.
__device__ __forceinline__ void issue_tile(const float* __restrict__ preds,
                                           const float* __restrict__ targs,
                                           int tile, uint32_t ldsP, uint32_t ldsT,
                                           int tid) {
    const char* gp = (const char*)preds + (size_t)tile * TILE_BYTES;
    const char* gt = (const char*)targs + (size_t)tile * TILE_BYTES;
    #pragma unroll
    for (int j = 0; j < 15; ++j) {
        int c = j * NTHREADS + tid;          // 0..3839
        bool isP = (c < CHUNKS);
        int cc = isP ? c : (c - CHUNKS);
        uint64_t ga = (uint64_t)(uintptr_t)((isP ? gp : gt) + (size_t)cc * 16);
        uint32_t la = (isP ? ldsP : ldsT) + (uint32_t)cc * 16u;
        asm volatile("global_load_async_to_lds_b128 %0, %1, off th:TH_LOAD_NT"
                     :: "v"(la), "v"(ga) : "memory");
    }
}

__global__ __launch_bounds__(NTHREADS)
void yolo_loss_partial(const float* __restrict__ preds,
                       const float* __restrict__ targs,
                       float* __restrict__ partial) {
    __shared__ float shP[2][TILE_FLOATS];   // double-buffered preds tile
    __shared__ float shT[2][TILE_FLOATS];   // double-buffered targets tile
    __shared__ float red[NTHREADS];

    const int tid = threadIdx.x;
    // Low 32 bits of a generic LDS pointer == byte offset in group segment
    // async-to-LDS takes this in a VGPR.
    const uint32_t ldsP0 = (uint32_t)(uintptr_t)(&shP[0][0]);
    const uint32_t ldsP1 = (uint32_t)(uintptr_t)(&shP[1][0]);
    const uint32_t ldsT0 = (uint32_t)(uintptr_t)(&shT[0][0]);
    const uint32_t ldsT1 = (uint32_t)(uintptr_t)(&shT[1][0]);

    float acc = 0.0f;

    int tile = blockIdx.x;          // every block has >= 6 tiles (6272/1024)
    int par = 0;
    if (tile < NTILES) issue_tile(preds, targs, tile, ldsP0, ldsT0, tid);

    while (tile < NTILES) {
        const int next = tile + gridDim.x;
        const bool hasNext = (next < NTILES);

        // Prefetch the next tile into the other buffer (its last readers
        // passed the trailing barrier in the previous iteration).
        if (hasNext) {
            issue_tile(preds, targs, next,
                       par ? ldsP0 : ldsP1, par ? ldsT0 : ldsT1, tid);
            // Async loads retire in issue order: ASYNCcnt <= 15 means the
            // current tile's 15 are done while the next 15 stay in flight.
            asm volatile("s_wait_asynccnt 0xf" ::: "memory");
        } else {
            asm volatile("s_wait_asynccnt 0x0" ::: "memory");
        }
        __syncthreads();            // all 8 waves' DMA for this buffer visible

        // --- per-cell loss (cell = tid of this tile) ---
        const float* p = (par ? &shP[1][0] : &shP[0][0]) + tid * FEAT;
        const float* t = (par ? &shT[1][0] : &shT[0][0]) + tid * FEAT;
        float pv[FEAT], tv[FEAT];
        #pragma unroll
        for (int k = 0; k < FEAT; ++k) { pv[k] = p[k]; tv[k] = t[k]; }

        float cell = 0.0f;
        // no-object confidence term over both boxes (all cells)
        {
            float d0 = pv[20] - tv[20];
            float d1 = pv[21] - tv[21];
            cell += NOOBJ_W * (d0 * d0 + d1 * d1);
        }
        // responsible box via IoU argmax (first max on tie, like jnp.argmax)
        float i0 = iou_f(pv[22], pv[23], pv[24], pv[25], tv[22], tv[23], tv[24], tv[25]);
        float i1 = iou_f(pv[26], pv[27], pv[28], pv[29], tv[26], tv[27], tv[28], tv[29]);
        bool top = (i1 > i0);
        // objectness mask: sum(t_probs) == 1.0 exactly (one-hot * obj)
        float sp = 0.0f;
        #pragma unroll
        for (int k = 0; k < 20; ++k) sp += tv[k];
        float m = (sp == 1.0f) ? 1.0f : 0.0f;
        // responsible-confidence term
        float pc = top ? pv[21] : pv[20];
        float tc = top ? tv[21] : tv[20];
        float dct = pc - tc;
        cell += (1.0f - NOOBJ_W) * m * dct * dct;
        // class-probability term
        float sq = 0.0f;
        #pragma unroll
        for (int k = 0; k < 20; ++k) { float d = pv[k] - tv[k]; sq += d * d; }
        cell += m * sq;
        // box terms (selects instead of dynamic indexing -> v_cndmask, no spills)
        float pbx = top ? pv[26] : pv[22], tbx = top ? tv[26] : tv[22];
        float pby = top ? pv[27] : pv[23], tby = top ? tv[27] : tv[23];
        float pbw = top ? pv[28] : pv[24], tbw = top ? tv[28] : tv[24];
        float pbh = top ? pv[29] : pv[25], tbh = top ? tv[29] : tv[25];
        float dx = pbx - tbx, dy = pby - tby;
        cell += m * (dx * dx + dy * dy);
        float dw = sqrtf(pbw) - sqrtf(tbw);
        float dh = sqrtf(pbh) - sqrtf(tbh);
        cell += m * (dw * dw + dh * dh);

        acc += cell;
        __syncthreads();   // readers done before this buffer is re-filled
        tile = next;
        par ^= 1;
    }

    // deterministic block tree-reduction
    red[tid] = acc;
    __syncthreads();
    #pragma unroll
    for (int s = NTHREADS / 2; s > 0; s >>= 1) {
        if (tid < s) red[tid] += red[tid + s];
        __syncthreads();
    }
    if (tid == 0) partial[blockIdx.x] = red[0];
}

__global__ __launch_bounds__(NTHREADS)
void yolo_loss_final(const float* __restrict__ partial, float* __restrict__ out) {
    __shared__ double red[NTHREADS];
    const int tid = threadIdx.x;
    double s = 0.0;
    for (int i = tid; i < NBLOCKS; i += NTHREADS) s += (double)partial[i];
    red[tid] = s;
    __syncthreads();
    #pragma unroll
    for (int k = NTHREADS / 2; k > 0; k >>= 1) {
        if (tid < k) red[tid] += red[tid + k];
        __syncthreads();
    }
    if (tid == 0) out[0] = (float)(red[0] * (1.0 / (double)NBATCH));
}

extern "C" void kernel_launch(void* const* d_in, const int* in_sizes, int n_in,
                              void* d_out, int out_size, void* d_ws, size_t ws_size,
                              hipStream_t stream) {
    const float* preds = (const float*)d_in[0];
    const float* targs = (const float*)d_in[1];
    float* partial = (float*)d_ws;          // NBLOCKS floats of scratch
    yolo_loss_partial<<<NBLOCKS, NTHREADS, 0, stream>>>(preds, targs, partial);
    yolo_loss_final<<<1, NTHREADS, 0, stream>>>(partial, (float*)d_out);
}